// NewRGCNLayer_8254927142973
// MI455X (gfx1250) — compile-verified
//
#include <hip/hip_runtime.h>
#include <hip/hip_bf16.h>

typedef __attribute__((ext_vector_type(2))) float v2f;
typedef __attribute__((ext_vector_type(8))) float v8f;

#define D 128

// ---------------------------------------------------------------------------
// Pass 1: edge scatter.  pre[dst] += h[src] + emb_rel[etype];  indeg[dst] = 1.
// One wave per edge (each lane owns 4 consecutive floats of the 128-wide row).
// pre (51.2 MB) is L2-resident on MI455X (192 MB L2) -> f32 atomics resolve
// in L2.  in_deg is only tested >0, so a racy plain store of 1.0 suffices.
// ---------------------------------------------------------------------------
__global__ __launch_bounds__(256) void rgcn_scatter_kernel(
    const float* __restrict__ h, const float* __restrict__ emb_rel,
    const int* __restrict__ src, const int* __restrict__ dst,
    const int* __restrict__ etype, float* __restrict__ pre,
    float* __restrict__ indeg, int E) {
  const int lane = threadIdx.x & 31;
  const int wave = (blockIdx.x * blockDim.x + threadIdx.x) >> 5;
  const int nwaves = (gridDim.x * blockDim.x) >> 5;
  for (int e = wave; e < E; e += nwaves) {
    const int s = src[e];
    const int d = dst[e];
    const int r = etype[e];
    const float4 hv = ((const float4*)(h + (size_t)s * D))[lane];
    const float4 ev = ((const float4*)(emb_rel + (size_t)r * D))[lane];
    float* pd = pre + (size_t)d * D + lane * 4;
    atomicAdd(pd + 0, hv.x + ev.x);
    atomicAdd(pd + 1, hv.y + ev.y);
    atomicAdd(pd + 2, hv.z + ev.z);
    atomicAdd(pd + 3, hv.w + ev.w);
    if (lane == 0) indeg[d] = 1.0f;
  }
}

// ---------------------------------------------------------------------------
// Pass 2: fused 4x GEMM (V_WMMA_F32_16X16X4_F32) + epilogue.
//   agg   = pre    @ W_neighbor
//   loopL = h      @ loop_weight        (in-degree  > 0)
//   loopE = h      @ evolve_loop_weight (in-degree == 0)
//   gateS = prev_h @ skip_connect_weight
//   out = relu( sig(gateS+b) * (agg*norm + loop) + (1-sig)*prev_h )
//
// Block = 8 waves; each wave owns a 16-row x 128-col output tile.  The block
// iterates the 8 n-tiles in lockstep: per n-tile, the 32 KB weight slab
// (4 matrices x 128K x 16N f32) is staged into LDS (bank-conflict-free), so
// B-fragments come off the DS pipe while the VMEM pipe streams the A rows
// (pre/h/prev_h), which is the real roofline term (~205 MB @ 23.3 TB/s).
// f32 WMMA is the right precision: memory-bound, so matrix throughput is not
// the limiter and we keep full f32 accuracy vs the reference.
// ---------------------------------------------------------------------------
__device__ __forceinline__ v8f wmma_f32(v2f a, v2f b, v8f c) {
  return __builtin_amdgcn_wmma_f32_16x16x4_f32(
      /*neg_a=*/false, a, /*neg_b=*/false, b,
      /*c_mod=*/(short)0, c, /*reuse_a=*/false, /*reuse_b=*/false);
}

__global__ __launch_bounds__(256) void rgcn_fused_gemm_kernel(
    const float* __restrict__ pre, const float* __restrict__ h,
    const float* __restrict__ prev_h, const float* __restrict__ Wn,
    const float* __restrict__ Wl, const float* __restrict__ We,
    const float* __restrict__ Ws, const float* __restrict__ bias,
    const float* __restrict__ norm, const float* __restrict__ indeg,
    float* __restrict__ out, int N) {
  __shared__ float wlds[4 * D * 16];          // 32 KB: [mat][k][16 cols]

  const int lane = threadIdx.x & 31;
  const int wave = threadIdx.x >> 5;
  int tile = blockIdx.x * 8 + wave;           // 16-row tile index
  const bool active = (tile * 16 < N);
  if (!active) tile = (N >> 4) - 1;           // clamp: recompute last tile,
                                              // stores suppressed below
  const int m0 = tile * 16;

  const int mr = m0 + (lane & 15);            // A-fragment row (same both halves)
  const int khalf = (lane >> 4) << 1;         // lanes 16-31 hold K+2,K+3
  const float* aPre = pre + (size_t)mr * D + khalf;
  const float* aH = h + (size_t)mr * D + khalf;
  const float* aP = prev_h + (size_t)mr * D + khalf;

  const int mbase = m0 + ((lane >> 4) << 3);  // C/D rows: mbase..mbase+7
  const int c = lane & 15;                    // this lane's column within n-tile

  // Hoist per-row epilogue scalars out of the n-tile loop.
  float nrm[8];
  bool deg[8];
  #pragma unroll
  for (int i = 0; i < 8; ++i) {
    nrm[i] = norm[mbase + i];
    deg[i] = indeg[mbase + i] > 0.0f;
  }

  for (int j = 0; j < 8; ++j) {               // n-tiles of 16 cols (block-synchronous)
    // ---- stage 32 KB weight slab for this n-tile into LDS ----
    __syncthreads();                          // previous tile's reads done
    for (int seg = threadIdx.x; seg < 4 * D; seg += 256) {
      const int mat = seg >> 7;               // 0..3
      const int kk = seg & (D - 1);           // 0..127
      const float* Wsrc = (mat == 0 ? Wn : mat == 1 ? Wl : mat == 2 ? We : Ws);
      const float4* s4 = (const float4*)(Wsrc + (size_t)kk * D + j * 16);
      float4* d4 = (float4*)(wlds + mat * (D * 16) + kk * 16);
      d4[0] = s4[0]; d4[1] = s4[1]; d4[2] = s4[2]; d4[3] = s4[3];
    }
    __syncthreads();

    // ---- 4 accumulators (one per weight matrix), K = 128 in steps of 4 ----
    v8f accA = {}, accL = {}, accE = {}, accS = {};
    for (int k = 0; k < D; k += 4) {
      const v2f a0 = *(const v2f*)(aPre + k);
      const v2f a1 = *(const v2f*)(aH + k);
      const v2f a2 = *(const v2f*)(aP + k);
      const int kb = (k + khalf) * 16 + c;    // LDS: rows k+khalf, k+khalf+1
      v2f b0, b1, b2, b3;
      b0.x = wlds[0 * (D * 16) + kb]; b0.y = wlds[0 * (D * 16) + kb + 16];
      b1.x = wlds[1 * (D * 16) + kb]; b1.y = wlds[1 * (D * 16) + kb + 16];
      b2.x = wlds[2 * (D * 16) + kb]; b2.y = wlds[2 * (D * 16) + kb + 16];
      b3.x = wlds[3 * (D * 16) + kb]; b3.y = wlds[3 * (D * 16) + kb + 16];
      accA = wmma_f32(a0, b0, accA);
      accL = wmma_f32(a1, b1, accL);
      accE = wmma_f32(a1, b2, accE);
      accS = wmma_f32(a2, b3, accS);
    }

    // ---- fused epilogue on accumulator registers ----
    if (active) {                             // wave-uniform: EXEC stays all-1s
      const int nc = j * 16 + c;
      const float bn = bias[nc];
      #pragma unroll
      for (int i = 0; i < 8; ++i) {
        const int m = mbase + i;
        const float loopv = deg[i] ? accL[i] : accE[i];
        const float node = accA[i] * nrm[i] + loopv;
        const float sw = 1.0f / (1.0f + __expf(-(accS[i] + bn)));
        const float v = sw * node + (1.0f - sw) * prev_h[(size_t)m * D + nc];
        out[(size_t)m * D + nc] = fmaxf(v, 0.0f);
      }
    }
  }
}

extern "C" void kernel_launch(void* const* d_in, const int* in_sizes, int n_in,
                              void* d_out, int out_size, void* d_ws, size_t ws_size,
                              hipStream_t stream) {
  const float* h = (const float*)d_in[0];
  const float* prev_h = (const float*)d_in[1];
  const float* emb_rel = (const float*)d_in[2];
  const float* norm = (const float*)d_in[3];
  const float* Wn = (const float*)d_in[4];
  const float* Wl = (const float*)d_in[5];
  const float* We = (const float*)d_in[6];
  const float* Ws = (const float*)d_in[7];
  const float* bias = (const float*)d_in[8];
  const int* src = (const int*)d_in[9];
  const int* dst = (const int*)d_in[10];
  const int* etype = (const int*)d_in[11];

  const int N = in_sizes[0] / D;   // 100000
  const int E = in_sizes[9];       // 600000

  // Workspace: pre [N*D f32] then indeg flags [N f32].
  float* pre = (float*)d_ws;
  float* indeg = pre + (size_t)N * D;

  hipMemsetAsync(pre, 0, (size_t)N * D * sizeof(float), stream);
  hipMemsetAsync(indeg, 0, (size_t)N * sizeof(float), stream);

  // Pass 1: edge scatter (memory/atomic bound; accumulator is L2-resident).
  rgcn_scatter_kernel<<<2048, 256, 0, stream>>>(h, emb_rel, src, dst, etype,
                                                pre, indeg, E);

  // Pass 2: fused GEMMs + epilogue.  16-row tiles, 8 waves per block.
  const int nTiles = (N + 15) / 16;            // N is a multiple of 16 here
  const int nBlocks = (nTiles + 7) / 8;
  rgcn_fused_gemm_kernel<<<nBlocks, 256, 0, stream>>>(
      pre, h, prev_h, Wn, Wl, We, Ws, bias, norm, indeg, (float*)d_out, N);
}